// PartialRadon_54408645705956
// MI455X (gfx1250) — compile-verified
//
#include <hip/hip_runtime.h>
#include <hip/hip_bf16.h>

#define A_ANG 180
#define NIMG  4
#define WDIM  256
#define SOUT  5

typedef __attribute__((ext_vector_type(2))) float v2f;
typedef __attribute__((ext_vector_type(8))) float v8f;

// Gather one pixel from the LDS-resident image with zero outside [0,256)^2.
__device__ __forceinline__ float fetch_lds(const float* __restrict__ smem, int y, int x) {
    const bool valid = ((unsigned)y < (unsigned)WDIM) && ((unsigned)x < (unsigned)WDIM);
    const int yc = min(max(y, 0), WDIM - 1);
    const int xc = min(max(x, 0), WDIM - 1);
    const float v = smem[yc * WDIM + xc];
    return valid ? v : 0.0f;
}

// Bilinear sample at (ix, iy) with zero padding — matches reference gather().
__device__ __forceinline__ float bilinear(const float* __restrict__ smem, float ix, float iy) {
    const float fx0 = floorf(ix);
    const float fy0 = floorf(iy);
    const int x0 = (int)fx0, y0 = (int)fy0;
    const int x1 = x0 + 1,   y1 = y0 + 1;
    const float wx1 = ix - fx0, wx0 = 1.0f - wx1;
    const float wy1 = iy - fy0, wy0 = 1.0f - wy1;
    float acc;
    acc  = fetch_lds(smem, y0, x0) * (wy0 * wx0);
    acc += fetch_lds(smem, y0, x1) * (wy0 * wx1);
    acc += fetch_lds(smem, y1, x0) * (wy1 * wx0);
    acc += fetch_lds(smem, y1, x1) * (wy1 * wx1);
    return acc;
}

__global__ __launch_bounds__(512) void radon_wmma_kernel(const float* __restrict__ x,
                                                         float* __restrict__ out) {
    // Whole 256x256 f32 image resident in LDS (256 KB of the WGP's 320 KB).
    __shared__ float smem[WDIM * WDIM];

    const int blk  = blockIdx.x;          // [0, 720)
    const int n    = blk / A_ANG;
    const int ang  = blk - n * A_ANG;
    const int tid  = threadIdx.x;
    const int lane = tid & 31;
    const int wave = tid >> 5;            // [0,16): one 16-wide w-tile per wave

    // Cooperative coalesced stage of image n into LDS (float4 = 128b moves).
    {
        const float4* __restrict__ src4 =
            reinterpret_cast<const float4*>(x + (size_t)n * WDIM * WDIM);
        float4* dst4 = reinterpret_cast<float4*>(smem);
        #pragma unroll 4
        for (int i = tid; i < (WDIM * WDIM) / 4; i += 512) {
            dst4[i] = src4[i];
        }
    }
    __syncthreads();

    const float theta = (float)ang * 0.017453292519943295f;  // a * pi/180
    float sn, cs;
    __sincosf(theta, &sn, &cs);

    // --- A-fragment (wts) lane geometry, per ISA 7.12.2 (32-bit A 16x4):
    //   lane<16 : M=lane,    VGPR0->K=k0+0, VGPR1->K=k0+1
    //   lane>=16: M=lane-16, VGPR0->K=k0+2, VGPR1->K=k0+3
    const int m    = lane & 15;            // matrix row (output s) / B column slot
    const int koff = (lane >> 4) << 1;     // 0 or 2 within the K=4 chunk
    // wts[s][h] = (|h-127.5| < seg[s]) / count[s], seg={1,32,64,96,128}, count={2,64,128,192,256}
    float thresh = 0.0f, invc = 0.0f;
    if (m < SOUT) {
        thresh = (m == 0) ? 1.0f : 32.0f * (float)m;
        invc   = (m == 0) ? 0.5f : 1.0f / (64.0f * (float)m);
    }

    // B-fragment: this lane owns column w = wave*16 + m.
    // Rotated-grid coords are affine in h:  ix(h) = ix0 + sn*h,  iy(h) = iy0 + cs*h
    //   ix0 =  128*cs*bw - 127.5*sn + 127.5
    //   iy0 = -128*sn*bw - 127.5*cs + 127.5
    const int   wcol = wave * 16 + m;
    const float bw   = ((float)wcol + 0.5f) * (2.0f / (float)WDIM) - 1.0f;
    const float ix0  = fmaf(128.0f * cs, bw, fmaf(-127.5f, sn, 127.5f));
    const float iy0  = fmaf(-128.0f * sn, bw, fmaf(-127.5f, cs, 127.5f));

    v8f acc = {};  // 16x16 f32 accumulator (rows 5..15 stay zero)

    #pragma unroll 4
    for (int chunk = 0; chunk < WDIM / 4; ++chunk) {
        const int   h0 = chunk * 4 + koff;
        const float hf = (float)h0;

        v2f afrag;
        afrag.x = (fabsf(hf        - 127.5f) < thresh) ? invc : 0.0f;
        afrag.y = (fabsf(hf + 1.0f - 127.5f) < thresh) ? invc : 0.0f;

        const float ix = fmaf(hf, sn, ix0);
        const float iy = fmaf(hf, cs, iy0);
        v2f bfrag;
        bfrag.x = bilinear(smem, ix,      iy);
        bfrag.y = bilinear(smem, ix + sn, iy + cs);

        // D = A(16x4,f32) * B(4x16,f32) + C  — full-precision matrix pipe.
        acc = __builtin_amdgcn_wmma_f32_16x16x4_f32(
            /*neg_a=*/false, afrag, /*neg_b=*/false, bfrag,
            /*c_mod=*/(short)0, acc, /*reuse_a=*/false, /*reuse_b=*/false);
    }

    // D layout: VGPR j, lanes 0-15 -> (M=j, N=lane). Rows 0..4 are the 5 ring outputs.
    if (lane < 16) {
        const int wout = wave * 16 + lane;
        const size_t stride_s = (size_t)WDIM * A_ANG;  // 256*180
        const size_t base = (((size_t)n * SOUT) * WDIM + wout) * A_ANG + ang;
        out[base + 0 * stride_s] = acc[0];
        out[base + 1 * stride_s] = acc[1];
        out[base + 2 * stride_s] = acc[2];
        out[base + 3 * stride_s] = acc[3];
        out[base + 4 * stride_s] = acc[4];
    }
}

extern "C" void kernel_launch(void* const* d_in, const int* in_sizes, int n_in,
                              void* d_out, int out_size, void* d_ws, size_t ws_size,
                              hipStream_t stream) {
    (void)in_sizes; (void)n_in; (void)out_size; (void)d_ws; (void)ws_size;
    const float* x = (const float*)d_in[0];
    float* out = (float*)d_out;
    dim3 grid(NIMG * A_ANG);   // one workgroup per (image, angle)
    dim3 block(512);           // 16 wave32: 4 waves per SIMD32, one w-tile per wave
    hipLaunchKernelGGL(radon_wmma_kernel, grid, block, 0, stream, x, out);
}